// Prediction_61692910240142
// MI455X (gfx1250) — compile-verified
//
#include <hip/hip_runtime.h>
#include <hip/hip_bf16.h>
#include <stdint.h>

// -------- constants matching the reference --------
#define SCORE_THR  0.05f
#define IOU_THR    0.5f
#define MAX_DET    300
#define K_PRE      512

typedef unsigned int u32x4 __attribute__((ext_vector_type(4)));
typedef int          i32x4 __attribute__((ext_vector_type(4)));
typedef int          i32x8 __attribute__((ext_vector_type(8)));

// Use the Tensor Data Mover when the toolchain exposes it.
#if defined(__HIP_DEVICE_COMPILE__) && defined(__gfx1250__) && __has_builtin(__builtin_amdgcn_tensor_load_to_lds)
  #define USE_TDM 1
#else
  #define USE_TDM 0
#endif

// ---------------- CDNA5 async global->LDS helpers ----------------
__device__ __forceinline__ void async_g2l_b128(uint32_t lds_byte_off,
                                               unsigned long long gaddr) {
#if defined(__gfx1250__)
  asm volatile("global_load_async_to_lds_b128 %0, %1, off"
               :: "v"(lds_byte_off), "v"(gaddr)
               : "memory");
#endif
}

__device__ __forceinline__ void wait_asynccnt0() {
#if defined(__gfx1250__)
  asm volatile("s_wait_asynccnt 0" ::: "memory");
#endif
}

__device__ __forceinline__ void wait_tensorcnt0() {
#if defined(__gfx1250__)
  asm volatile("s_wait_tensorcnt 0" ::: "memory");
#endif
}

// =================================================================
// Kernel 1: per-row max/argmax over C=80 class scores.
// 128 rows per block, tile staged into LDS.
//  - TDM path: one tensor_load_to_lds DMAs the 40KB tile; TDM padding
//    (1 DWORD per 16) gives an 85-DWORD row stride -> conflict-free LDS.
//  - Fallback: per-lane global_load_async_to_lds_b128 (84-float stride).
// =================================================================
__global__ __launch_bounds__(128)
void score_kernel(const float* __restrict__ cls,
                  float* __restrict__ scores,
                  int*   __restrict__ labels,
                  long totalRows, int C) {
  const int  t    = threadIdx.x;             // 0..127
  const long row0 = (long)blockIdx.x * 128;

#if USE_TDM
  if (C == 80 && row0 + 128 <= totalRows) {
    // 10240 elements + 640 pad DWORDs = 43520 B
    __shared__ __align__(16) float tile[128 * 85];
    const uint32_t lbase = (uint32_t)(uintptr_t)(&tile[0]);
    const unsigned long long g =
        (unsigned long long)(uintptr_t)(cls + row0 * 80);

    if (t < 32) {                             // one wave issues the DMA
      u32x4 g0;
      g0.x = 1u;                              // count=1, user descriptor
      g0.y = lbase;                           // lds_addr (bytes)
      g0.z = (unsigned)(g & 0xFFFFFFFFull);   // global_addr[31:0]
      g0.w = (unsigned)((g >> 32) & 0x1FFFFFFull) | (2u << 30); // [56:32] | type=2

      i32x8 g1;
      // data_size=4B(2), pad_enable, pad_interval=16 DWORDs(3), pad_amount=1 DWORD(0)
      g1[0] = (int)((2u << 16) | (1u << 20) | (3u << 22));
      g1[1] = (int)((10240u & 0xFFFFu) << 16);      // tensor_dim0[15:0]
      g1[2] = (int)((10240u >> 16) | (1u << 16));   // tensor_dim0[31:16] | tensor_dim1[15:0]=1
      g1[3] = (int)(10240u << 16);                  // tensor_dim1[31:16]=0 | tile_dim0=10240
      g1[4] = 0;                                    // tile_dim1=0 (1-D), tile_dim2=0
      g1[5] = 10240;                                // tensor_dim0_stride[31:0]
      g1[6] = 0;
      g1[7] = 0;

      i32x4 z4 = {0, 0, 0, 0};
#if __clang_major__ >= 23
      i32x8 z8 = {0, 0, 0, 0, 0, 0, 0, 0};
      __builtin_amdgcn_tensor_load_to_lds(g0, g1, z4, z4, z8, 0);
#else
      __builtin_amdgcn_tensor_load_to_lds(g0, g1, z4, z4, 0);
#endif
      wait_tensorcnt0();
    }
    __syncthreads();

    // row t lives at DWORD 85*t; element c at +17*(c/16) + c%16
    const float* rp = &tile[t * 85];
    float best = -3.402823466e38f;
    int   bi   = 0;
    #pragma unroll
    for (int gch = 0; gch < 5; ++gch) {
      #pragma unroll
      for (int cc = 0; cc < 16; ++cc) {
        float v = rp[gch * 17 + cc];
        int   c = gch * 16 + cc;
        if (v > best) { best = v; bi = c; }
      }
    }
    scores[row0 + t] = best;
    labels[row0 + t] = bi;
    return;
  }
#else
  if (C == 80 && row0 + 128 <= totalRows) {
    __shared__ __align__(16) float tile[128 * 84];   // 84-float padded rows
    const uint32_t lbase = (uint32_t)(uintptr_t)(&tile[0]);
    const float* src = cls + row0 * 80;

    #pragma unroll
    for (int k = 0; k < 20; ++k) {
      const int f    = k * 128 + t;          // float4 chunk id (0..2559)
      const int rrow = f / 20;
      const int cc   = f % 20;
      const uint32_t loff = lbase + (uint32_t)(rrow * 336 + cc * 16);
      const unsigned long long g = (unsigned long long)(uintptr_t)(src + (long)f * 4);
      async_g2l_b128(loff, g);
    }
    wait_asynccnt0();
    __syncthreads();

    const float4* rp = (const float4*)(&tile[t * 84]);
    float best = -3.402823466e38f;
    int   bi   = 0;
    #pragma unroll
    for (int c = 0; c < 20; ++c) {
      float4 q = rp[c];
      if (q.x > best) { best = q.x; bi = c * 4 + 0; }
      if (q.y > best) { best = q.y; bi = c * 4 + 1; }
      if (q.z > best) { best = q.z; bi = c * 4 + 2; }
      if (q.w > best) { best = q.w; bi = c * 4 + 3; }
    }
    scores[row0 + t] = best;
    labels[row0 + t] = bi;
    return;
  }
#endif
  // generic / tail path (direct loads)
  {
    const long row = row0 + t;
    if (row < totalRows) {
      const float* r = cls + row * (long)C;
      float best = -3.402823466e38f;
      int   bi   = 0;
      for (int c = 0; c < C; ++c) {
        float v = r[c];
        if (v > best) { best = v; bi = c; }
      }
      scores[row] = best;
      labels[row] = bi;
    }
  }
}

// =================================================================
// Kernel 2: exact top-512 per batch via 4-pass byte radix-select on the
// float bit pattern (monotone for positive floats), then compaction.
// One 1024-thread block per batch.
// =================================================================
__global__ __launch_bounds__(1024)
void topk_kernel(const float* __restrict__ scores,
                 unsigned* __restrict__ ckey,
                 int*      __restrict__ cidx,
                 int N) {
  const int b = blockIdx.x;
  const int t = threadIdx.x;
  const float* s = scores + (size_t)b * N;

  __shared__ unsigned hist[256];
  __shared__ unsigned sh_prefix;
  __shared__ int      sh_need;
  __shared__ unsigned sh_greater;
  __shared__ int      sh_cntG, sh_cntE;

  if (t == 0) { sh_prefix = 0u; sh_need = K_PRE; sh_greater = 0u; }
  __syncthreads();

  for (int pass = 3; pass >= 0; --pass) {
    for (int i = t; i < 256; i += blockDim.x) hist[i] = 0u;
    __syncthreads();
    const unsigned pre   = sh_prefix;
    const int      shift = pass * 8;
    for (int n = t; n < N; n += blockDim.x) {
      float v = s[n];
      if (v > SCORE_THR) {
        unsigned key = __float_as_uint(v);
        bool match = (pass == 3) || ((key >> (shift + 8)) == pre);
        if (match) atomicAdd(&hist[(key >> shift) & 255u], 1u);
      }
    }
    __syncthreads();
    if (t == 0) {
      int need = sh_need;
      int acc  = 0;
      int sel  = -1;
      for (int bkt = 255; bkt >= 0; --bkt) {
        int c = (int)hist[bkt];
        if (acc + c >= need) { sel = bkt; break; }
        acc += c;
      }
      if (sel < 0) { sel = 0; acc -= (int)hist[0]; }  // fewer than `need` remain
      sh_greater += (unsigned)acc;
      sh_need     = need - acc;       // invariant: sh_greater + sh_need == K_PRE
      sh_prefix   = (pre << 8) | (unsigned)sel;
    }
    __syncthreads();
  }

  const unsigned T       = sh_prefix;           // exact 512th-largest key
  const int      nGreat  = (int)sh_greater;     // strictly greater than T
  const int      needEq  = sh_need;             // how many == T to take
  if (t == 0) { sh_cntG = 0; sh_cntE = 0; }
  __syncthreads();

  unsigned* bk = ckey + (size_t)b * K_PRE;
  int*      bi = cidx + (size_t)b * K_PRE;
  for (int n = t; n < N; n += blockDim.x) {
    float v = s[n];
    if (v > SCORE_THR) {
      unsigned key = __float_as_uint(v);
      if (key > T) {
        int p = atomicAdd(&sh_cntG, 1);
        bk[p] = key; bi[p] = n;
      } else if (key == T) {
        int e = atomicAdd(&sh_cntE, 1);
        if (e < needEq) { bk[nGreat + e] = key; bi[nGreat + e] = n; }
      }
    }
  }
  __syncthreads();
  const int eq    = (sh_cntE < needEq) ? sh_cntE : needEq;
  const int total = nGreat + eq;
  for (int p = total + t; p < K_PRE; p += blockDim.x) { bk[p] = 0u; bi[p] = 0; }
}

// =================================================================
// Kernel 3: per batch (512 threads): candidates staged via async-to-LDS,
// bitonic sort (desc score, asc index), decode+clip, greedy NMS
// (reference semantics), emit top-300 padded with -1.
// Output layout: [boxes | scores | labels].
// =================================================================
__global__ __launch_bounds__(512)
void nms_kernel(const float* __restrict__ anchors,
                const float* __restrict__ reg,
                const float* __restrict__ sizes,
                const int*   __restrict__ labels,
                const unsigned* __restrict__ ckey,
                const int*      __restrict__ cidx,
                float* __restrict__ out,
                int N, int B) {
  const int b = blockIdx.x;
  const int t = threadIdx.x;            // 0..511

  __shared__ __align__(16) unsigned s_key[K_PRE];
  __shared__ __align__(16) int      s_idx[K_PRE];
  __shared__ unsigned long long sk[K_PRE];
  __shared__ float bx0[K_PRE], by0[K_PRE], bx1[K_PRE], by1[K_PRE];
  __shared__ unsigned char keep[K_PRE];
  __shared__ int lab[K_PRE];
  __shared__ int psum[K_PRE];

  // stage candidate arrays through the async global->LDS path (ASYNCcnt)
  {
    const uint32_t kb = (uint32_t)(uintptr_t)(&s_key[0]);
    const uint32_t ib = (uint32_t)(uintptr_t)(&s_idx[0]);
    if (t < 128) {            // waves 0..3: 128 x 16B = 2KB of keys
      async_g2l_b128(kb + (uint32_t)t * 16u,
                     (unsigned long long)(uintptr_t)(ckey + (size_t)b * K_PRE + (size_t)t * 4));
    } else if (t < 256) {     // waves 4..7: 2KB of indices
      const int u = t - 128;
      async_g2l_b128(ib + (uint32_t)u * 16u,
                     (unsigned long long)(uintptr_t)(cidx + (size_t)b * K_PRE + (size_t)u * 4));
    }
    wait_asynccnt0();
    __syncthreads();
  }

  {
    unsigned key = s_key[t];
    unsigned idx = (unsigned)s_idx[t];
    // composite: higher score first; among equal scores, smaller index first.
    sk[t] = ((unsigned long long)key << 32) | (unsigned)(~idx);
  }
  __syncthreads();

  // bitonic sort, overall descending
  for (int k = 2; k <= K_PRE; k <<= 1) {
    for (int j = k >> 1; j > 0; j >>= 1) {
      __syncthreads();
      int i = t ^ j;
      if (i > t) {
        unsigned long long a = sk[t], c = sk[i];
        bool desc = ((t & k) == 0);
        if (desc ? (a < c) : (a > c)) { sk[t] = c; sk[i] = a; }
      }
    }
  }
  __syncthreads();

  const unsigned long long v = sk[t];
  const unsigned k32  = (unsigned)(v >> 32);
  const int      idx2 = (int)(~(unsigned)v);
  const bool valid = (k32 != 0u);
  const float score = __uint_as_float(k32);

  float x1 = 0.f, y1 = 0.f, x2 = 0.f, y2 = 0.f;
  int   lb = -1;
  if (valid) {
    const float* a = anchors + ((size_t)b * N + idx2) * 4;
    const float* r = reg     + ((size_t)b * N + idx2) * 4;
    float aw = a[2] - a[0], ah = a[3] - a[1];
    float cx = a[0] + 0.5f * aw, cy = a[1] + 0.5f * ah;
    float dx = r[0] * 0.1f, dy = r[1] * 0.1f;
    float dw = r[2] * 0.2f, dh = r[3] * 0.2f;
    float pcx = cx + dx * aw, pcy = cy + dy * ah;
    float pw = expf(dw) * aw, ph = expf(dh) * ah;
    x1 = pcx - 0.5f * pw; y1 = pcy - 0.5f * ph;
    x2 = pcx + 0.5f * pw; y2 = pcy + 0.5f * ph;
    const float W = sizes[b * 2 + 1], H = sizes[b * 2 + 0];
    x1 = fminf(fmaxf(x1, 0.f), W); x2 = fminf(fmaxf(x2, 0.f), W);
    y1 = fminf(fmaxf(y1, 0.f), H); y2 = fminf(fmaxf(y2, 0.f), H);
    lb = labels[(size_t)b * N + idx2];
  }
  bx0[t] = x1; by0[t] = y1; bx1[t] = x2; by1[t] = y2;
  keep[t] = valid ? 1 : 0;
  lab[t]  = lb;
  __syncthreads();

  // greedy NMS, identical semantics to the reference fori_loop
  const float areaT = (x2 - x1) * (y2 - y1);
  for (int i = 0; i < K_PRE; ++i) {
    bool ki = (keep[i] != 0);
    if (ki && t > i && keep[t]) {
      float ax1 = bx0[i], ay1 = by0[i], ax2 = bx1[i], ay2 = by1[i];
      float areaI = (ax2 - ax1) * (ay2 - ay1);
      float lx = fmaxf(ax1, x1), ly = fmaxf(ay1, y1);
      float rx = fminf(ax2, x2), ry = fminf(ay2, y2);
      float iw = fmaxf(rx - lx, 0.f), ih = fmaxf(ry - ly, 0.f);
      float inter = iw * ih;
      float iou = inter / (areaI + areaT - inter + 1e-9f);
      if (iou > IOU_THR) keep[t] = 0;
    }
    __syncthreads();
  }

  // inclusive prefix sum of keep flags (Hillis-Steele)
  psum[t] = (int)keep[t];
  __syncthreads();
  for (int off = 1; off < K_PRE; off <<= 1) {
    int mine = psum[t];
    int add  = (t >= off) ? psum[t - off] : 0;
    __syncthreads();
    psum[t] = mine + add;
    __syncthreads();
  }

  int total = psum[K_PRE - 1];
  if (total > MAX_DET) total = MAX_DET;

  float* outB = out;                                   // [B,300,4]
  float* outS = out + (size_t)B * MAX_DET * 4;         // [B,300,1]
  float* outL = outS + (size_t)B * MAX_DET;            // [B,300,1]

  if (keep[t]) {
    int rnk = psum[t] - 1;
    if (rnk < MAX_DET) {
      float* ob = outB + ((size_t)b * MAX_DET + rnk) * 4;
      ob[0] = x1; ob[1] = y1; ob[2] = x2; ob[3] = y2;
      outS[(size_t)b * MAX_DET + rnk] = score;
      outL[(size_t)b * MAX_DET + rnk] = (float)lab[t];
    }
  }
  if (t < MAX_DET && t >= total) {
    float* ob = outB + ((size_t)b * MAX_DET + t) * 4;
    ob[0] = -1.f; ob[1] = -1.f; ob[2] = -1.f; ob[3] = -1.f;
    outS[(size_t)b * MAX_DET + t] = -1.f;
    outL[(size_t)b * MAX_DET + t] = -1.f;
  }
}

// =================================================================
extern "C" void kernel_launch(void* const* d_in, const int* in_sizes, int n_in,
                              void* d_out, int out_size, void* d_ws, size_t ws_size,
                              hipStream_t stream) {
  const float* anchors = (const float*)d_in[0];
  const float* reg     = (const float*)d_in[1];
  const float* cls     = (const float*)d_in[2];
  const float* sizes   = (const float*)d_in[3];

  const int  B  = in_sizes[3] / 2;                   // sizes is [B,2]
  const long BN = (long)in_sizes[0] / 4;             // anchors is [B,N,4]
  const int  N  = (int)(BN / B);
  const int  C  = (int)((long)in_sizes[2] / BN);     // cls is [B,N,C]

  // workspace layout: scores[BN] f32 | labels[BN] i32 | ckey[B*512] u32 | cidx[B*512] i32
  float*    ws_scores = (float*)d_ws;
  int*      ws_labels = (int*)(ws_scores + BN);
  unsigned* ws_ckey   = (unsigned*)(ws_labels + BN);
  int*      ws_cidx   = (int*)(ws_ckey + (size_t)B * K_PRE);

  const long totalRows = BN;
  const int  blocks1   = (int)((totalRows + 127) / 128);

  score_kernel<<<blocks1, 128, 0, stream>>>(cls, ws_scores, ws_labels, totalRows, C);
  topk_kernel<<<B, 1024, 0, stream>>>(ws_scores, ws_ckey, ws_cidx, N);
  nms_kernel<<<B, K_PRE, 0, stream>>>(anchors, reg, sizes, ws_labels,
                                      ws_ckey, ws_cidx, (float*)d_out, N, B);
}